// HybridDynamics_17875653886643
// MI455X (gfx1250) — compile-verified
//
#include <hip/hip_runtime.h>

// ---------------------------------------------------------------------------
// HybridDynamics on MI455X (gfx1250, wave32).
//
// Memory-bound streaming kernel (67 MB @ 23.3 TB/s ~= 2.9 us floor).
// The 4 tiny GEMMs run on the matrix pipe via V_WMMA_F32_16X16X4_F32
// (full fp32 -- no precision downcast, since bandwidth, not math, is the
// limit). Layout: neurons on M, batch on N, so the Gaussian activation is
// element-wise on the D accumulator and inter-layer repacking is a
// half-wave swap (ds_swizzle SWAPX16) feeding the B operand's K=2,3 rows.
//
// Assumed f32 operand layouts (CDNA5 ISA 7.12.2):
//   A (16x4):  lanes 0-15 hold M=lane, VGPR0=K0, VGPR1=K1;
//              lanes 16-31 hold M=lane-16, VGPR0=K2, VGPR1=K3.
//   B (4x16):  lanes index N; VGPR0={K0 | K2}, VGPR1={K1 | K3} per lane half.
//   C/D (16x16): VGPR v: lanes 0-15 -> M=v, lanes 16-31 -> M=v+8.
// Weight rows >= neuron count and K-padding are zero-filled, so the unused
// upper-half accumulator rows are exactly zero and never leak into B.
//
// The half-wave swaps are raw `ds_swizzle_b32 offset:0x401f` (SWAPX16:
// and=0x1F or=0 xor=0x10, group-of-32 mode) in *volatile* inline asm so the
// compiler cannot predicate them into divergent regions (ds permutes read
// zero from EXEC-disabled lanes, which would corrupt the repack -- this is
// exactly what the previous round's __shfl_xor lowering did).
// ---------------------------------------------------------------------------

typedef float v2f __attribute__((ext_vector_type(2)));
typedef float v8f __attribute__((ext_vector_type(8)));

// Swap wave halves for two values at once: out[lane] = in[lane ^ 16].
// Executed unconditionally (volatile) with EXEC all-1s; single dscnt drain.
__device__ __forceinline__ void swap16_pair(float a, float b,
                                            float& oa, float& ob) {
    asm volatile("ds_swizzle_b32 %0, %2 offset:0x401f\n\t"
                 "ds_swizzle_b32 %1, %3 offset:0x401f\n\t"
                 "s_wait_dscnt 0x0"
                 : "=&v"(oa), "=&v"(ob)
                 : "v"(a), "v"(b));
}

__device__ __forceinline__ v8f wmma4(v2f a, v2f b, v8f c) {
    // D = A(16x4 f32) x B(4x16 f32) + C(16x16 f32)
    return __builtin_amdgcn_wmma_f32_16x16x4_f32(
        /*neg_a=*/false, a, /*neg_b=*/false, b,
        /*c_mod=*/(short)0, c, /*reuse_a=*/false, /*reuse_b=*/false);
}

__global__ __launch_bounds__(256) void hybrid_dynamics_wmma(
    const float* __restrict__ x,
    const float* __restrict__ w1, const float* __restrict__ b1,
    const float* __restrict__ w2, const float* __restrict__ b2,
    const float* __restrict__ w3, const float* __restrict__ b3,
    const float* __restrict__ w4, const float* __restrict__ b4,
    const float* __restrict__ gamma,
    float* __restrict__ out, int n)
{
    const int  lane = threadIdx.x & 31;
    // Wave-uniform loop controls (readfirstlane -> SGPR) so the grid-stride
    // loop is a scalar branch and EXEC stays all-1s at every WMMA.
    const int  waveId =
        __builtin_amdgcn_readfirstlane((blockIdx.x * blockDim.x + threadIdx.x) >> 5);
    const int  nWaves = (gridDim.x * blockDim.x) >> 5;
    const int  m      = lane & 15;       // neuron row this lane feeds in A
    const bool lo     = lane < 16;       // lower lane half (K=0,1 / M=v rows)

    // ---- hoisted A-operands (weights). Zero-padded rows/K. ----
    // Layer 1: W1 is 5x2 -> K=2,3 pad = 0 (upper half zero).
    v2f A1;
    A1.x = (lo && m < 5) ? w1[m * 2 + 0] : 0.f;
    A1.y = (lo && m < 5) ? w1[m * 2 + 1] : 0.f;
    // Layers 2/3: W is 5x5 -> K=0..4; first WMMA covers K0-3, second K4 only.
    v2f A2a, A2b, A3a, A3b;
    A2a.x = (m < 5) ? (lo ? w2[m * 5 + 0] : w2[m * 5 + 2]) : 0.f;
    A2a.y = (m < 5) ? (lo ? w2[m * 5 + 1] : w2[m * 5 + 3]) : 0.f;
    A2b.x = (lo && m < 5) ? w2[m * 5 + 4] : 0.f;  A2b.y = 0.f;
    A3a.x = (m < 5) ? (lo ? w3[m * 5 + 0] : w3[m * 5 + 2]) : 0.f;
    A3a.y = (m < 5) ? (lo ? w3[m * 5 + 1] : w3[m * 5 + 3]) : 0.f;
    A3b.x = (lo && m < 5) ? w3[m * 5 + 4] : 0.f;  A3b.y = 0.f;
    // Layer 4: W4 is 2x5.
    v2f A4a, A4b;
    A4a.x = (m < 2) ? (lo ? w4[m * 5 + 0] : w4[m * 5 + 2]) : 0.f;
    A4a.y = (m < 2) ? (lo ? w4[m * 5 + 1] : w4[m * 5 + 3]) : 0.f;
    A4b.x = (lo && m < 2) ? w4[m * 5 + 4] : 0.f;  A4b.y = 0.f;

    // ---- hoisted bias C-matrices (broadcast across N; zero on M>=5 rows) ----
    v8f C1 = {}, C2 = {}, C3 = {}, C4 = {};
#pragma unroll
    for (int j = 0; j < 5; ++j) {
        C1[j] = lo ? b1[j] : 0.f;
        C2[j] = lo ? b2[j] : 0.f;
        C3[j] = lo ? b3[j] : 0.f;
    }
    C4[0] = lo ? b4[0] : 0.f;
    C4[1] = lo ? b4[1] : 0.f;

    const float g0 = gamma[0], g1 = gamma[1];

    // One 16-state tile: B-operand inputs (bx,by) pre-masked so upper half
    // (K=2,3 pad of layer 1) is zero. Returns dx0/dx1 in lanes 0-15.
    auto tileForward = [&](float bx, float by, float& o0, float& o1) {
        v2f B;  B.x = bx;  B.y = by;
        v8f h = wmma4(A1, B, C1);                       // layer 1 (K=2)

#pragma unroll
        for (int L = 0; L < 3; ++L) {                   // layers 2, 3, 4
            // Gaussian activation on D rows 0..4 (upper half holds zeros,
            // whose exp()=1 garbage is masked out during repack below).
            float s0 = __expf(-h[0] * h[0]);
            float s1 = __expf(-h[1] * h[1]);
            float s2 = __expf(-h[2] * h[2]);
            float s3 = __expf(-h[3] * h[3]);
            float s4 = __expf(-h[4] * h[4]);
            // Repack D -> B (K=5): rows 2,3 cross from lower lane half into
            // the upper half's K=2,3 slots via an unconditional SWAPX16.
            float w2s, w3s;
            swap16_pair(s2, s3, w2s, w3s);
            v2f Ba, Bb;
            Ba.x = lo ? s0 : w2s;                       // {K0 | K2}
            Ba.y = lo ? s1 : w3s;                       // {K1 | K3}
            Bb.x = lo ? s4 : 0.f;                       // {K4 | 0 }
            Bb.y = 0.f;
            if      (L == 0) { h = wmma4(A2a, Ba, C2); h = wmma4(A2b, Bb, h); }
            else if (L == 1) { h = wmma4(A3a, Ba, C3); h = wmma4(A3b, Bb, h); }
            else             { h = wmma4(A4a, Ba, C4); h = wmma4(A4b, Bb, h); }
        }
        o0 = h[0];   // dx0 for state N=lane (valid in lanes 0-15)
        o1 = h[1];   // dx1
    };

    // ---- grid-stride over 32-state groups (2 tiles per wave iteration) ----
    for (int base = waveId * 32; base < n; base += nWaves * 32) {
        const float2 xv =
            *reinterpret_cast<const float2*>(x + 2 * (base + lane));

        const int nb = base + nWaves * 32;              // prefetch next chunk
        if (nb < n) __builtin_prefetch(x + 2 * (nb + lane), 0, 1);

        // Tile 0: states base+0..15 already in lanes 0-15.
        // Tile 1: states base+16..31 swapped down into lanes 0-15.
        float sx, sy;
        swap16_pair(xv.x, xv.y, sx, sy);
        const float t0x = lo ? xv.x : 0.f;
        const float t0y = lo ? xv.y : 0.f;
        const float t1x = lo ? sx : 0.f;
        const float t1y = lo ? sy : 0.f;

        float d00, d01, d10, d11;
        tileForward(t0x, t0y, d00, d01);
        tileForward(t1x, t1y, d10, d11);

        // Reassemble per-lane float2: lanes 16-31 pull tile 1 back up.
        float u0, u1;
        swap16_pair(d10, d11, u0, u1);
        const float o0 = lo ? d00 : u0;
        const float o1 = lo ? d01 : u1;

        float2 ov;
        ov.x = fmaf(g0, xv.x, o0);                      // dx + gamma * x
        ov.y = fmaf(g1, xv.y, o1);
        *reinterpret_cast<float2*>(out + 2 * (base + lane)) = ov;
    }
}

extern "C" void kernel_launch(void* const* d_in, const int* in_sizes, int n_in,
                              void* d_out, int out_size, void* d_ws, size_t ws_size,
                              hipStream_t stream) {
    // setup_inputs order: t, x, w1, b1, w2, b2, w3, b3, w4, b4, gamma
    const float* x     = (const float*)d_in[1];
    const float* w1    = (const float*)d_in[2];
    const float* b1    = (const float*)d_in[3];
    const float* w2    = (const float*)d_in[4];
    const float* b2    = (const float*)d_in[5];
    const float* w3    = (const float*)d_in[6];
    const float* b3    = (const float*)d_in[7];
    const float* w4    = (const float*)d_in[8];
    const float* b4    = (const float*)d_in[9];
    const float* gamma = (const float*)d_in[10];
    float*       out   = (float*)d_out;

    const int n = in_sizes[1] / 2;                     // number of states (B)

    const int threads = 256;                           // 8 waves per block
    // ~2 loop iterations per wave: B/32 wave-iterations total.
    int groups = (n + 31) / 32;                        // 32 states per group
    int waves  = (groups + 1) / 2;
    int blocks = (waves * 32 + threads - 1) / threads;
    if (blocks < 1) blocks = 1;

    hybrid_dynamics_wmma<<<blocks, threads, 0, stream>>>(
        x, w1, b1, w2, b2, w3, b3, w4, b4, gamma, out, n);
}